// PadMaskedSequence_1700807049705
// MI455X (gfx1250) — compile-verified
//
#include <hip/hip_runtime.h>

typedef float float4v __attribute__((ext_vector_type(4)));

// ---------------------------------------------------------------------------
// Kernel 1: per-batch mask prefix scan. Two-level scan:
//   level 0: wave32 inclusive scan via __shfl_up (lane-permute path, no LDS)
//   level 1: wave 0 scans the 32 wave totals (one LDS slot per wave)
// Only 2 barriers / 2 LDS touches per thread. One 1024-thread block per batch
// row; each thread owns 4 mask bytes (requires T <= 4096, T % 4 == 0).
// Produces srcIdx[b*T + j] = original index of the j-th kept row, lens[b].
// ---------------------------------------------------------------------------
__global__ __launch_bounds__(1024)
void pms_scan_kernel(const unsigned char* __restrict__ mask,
                     int* __restrict__ srcIdx,
                     int* __restrict__ lens,
                     float* __restrict__ lensOut,
                     int T)
{
    const int b    = blockIdx.x;
    const int tid  = threadIdx.x;
    const int lane = tid & 31;
    const int wid  = tid >> 5;                 // 32 waves of 32 lanes
    __shared__ int waveTot[32];

    // 4 bool bytes per thread as one aligned dword load.
    unsigned int mword = 0u;
    const int t0 = tid * 4;
    if (t0 < T) {
        mword = *reinterpret_cast<const unsigned int*>(mask + (size_t)b * T + t0);
    }
    const int m0 = (int)( mword        & 1u);
    const int m1 = (int)((mword >>  8) & 1u);
    const int m2 = (int)((mword >> 16) & 1u);
    const int m3 = (int)((mword >> 24) & 1u);
    const int s  = m0 + m1 + m2 + m3;

    // Wave32 inclusive scan of per-thread sums (5 shuffle steps).
    int incl = s;
    #pragma unroll
    for (int off = 1; off < 32; off <<= 1) {
        int n = __shfl_up(incl, off, 32);
        if (lane >= off) incl += n;
    }
    if (lane == 31) waveTot[wid] = incl;       // wave total
    __syncthreads();

    // Wave 0 turns the 32 wave totals into inclusive totals.
    if (wid == 0) {
        int w = waveTot[lane];
        #pragma unroll
        for (int off = 1; off < 32; off <<= 1) {
            int n = __shfl_up(w, off, 32);
            if (lane >= off) w += n;
        }
        waveTot[lane] = w;
    }
    __syncthreads();

    const int waveBase = (wid > 0) ? waveTot[wid - 1] : 0;
    int pos = waveBase + incl - s;             // exclusive prefix for this chunk

    int* rowIdx = srcIdx + (size_t)b * T;
    if (m0) rowIdx[pos++] = t0 + 0;
    if (m1) rowIdx[pos++] = t0 + 1;
    if (m2) rowIdx[pos++] = t0 + 2;
    if (m3) rowIdx[pos++] = t0 + 3;

    if (tid == 0) {
        const int total = waveTot[31];         // grand total
        lens[b]    = total;                    // int scratch for gather kernel
        lensOut[b] = (float)total;             // reference output #2 (float d_out)
    }
}

// ---------------------------------------------------------------------------
// Kernel 2 (pow2 fast path): output-stationary gather + pad, one thread per
// output float4. Index math is pure shifts/masks. b128 non-temporal
// loads/stores: x and out are each touched exactly once; keep L2 for the hot
// srcIdx/lens metadata instead.
// ---------------------------------------------------------------------------
__global__ __launch_bounds__(256)
void pms_gather_pow2(const float4v* __restrict__ x4,
                     const int* __restrict__ srcIdx,
                     const int* __restrict__ lens,
                     float4v* __restrict__ out4,
                     unsigned int shiftD4, unsigned int maskD4,
                     unsigned int shiftT,  unsigned int maskT,
                     unsigned int total4)
{
    const unsigned int g = blockIdx.x * blockDim.x + threadIdx.x;
    if (g >= total4) return;

    const unsigned int lane = g &  maskD4;     // float4 lane within the row
    const unsigned int row  = g >> shiftD4;    // global output row (b*T + j)
    const unsigned int b    = row >> shiftT;
    const unsigned int j    = row &  maskT;

    float4v v = (float4v)0.0f;
    if ((int)j < lens[b]) {
        const int s = srcIdx[row];             // original row index for slot j
        const float4v* src = x4 + (((size_t)row - j + (unsigned)s) << shiftD4) + lane;
        v = __builtin_nontemporal_load(src);   // global_load_b128, TH=NT
    }
    __builtin_nontemporal_store(v, out4 + g);  // global_store_b128, TH=NT
}

// Generic fallback (non-pow2 shapes): same logic with divisions.
__global__ __launch_bounds__(256)
void pms_gather_div(const float4v* __restrict__ x4,
                    const int* __restrict__ srcIdx,
                    const int* __restrict__ lens,
                    float4v* __restrict__ out4,
                    unsigned int T, unsigned int D4, unsigned int total4)
{
    const unsigned int g = blockIdx.x * blockDim.x + threadIdx.x;
    if (g >= total4) return;

    const unsigned int lane = g % D4;
    const unsigned int row  = g / D4;
    const unsigned int b    = row / T;
    const unsigned int j    = row % T;

    float4v v = (float4v)0.0f;
    if ((int)j < lens[b]) {
        const int s = srcIdx[row];
        const float4v* src = x4 + ((size_t)b * T + (size_t)s) * D4 + lane;
        v = __builtin_nontemporal_load(src);
    }
    __builtin_nontemporal_store(v, out4 + g);
}

// ---------------------------------------------------------------------------
static inline unsigned int ilog2u(unsigned int v) {
    unsigned int r = 0;
    while ((1u << r) < v) ++r;
    return r;
}

extern "C" void kernel_launch(void* const* d_in, const int* in_sizes, int n_in,
                              void* d_out, int out_size, void* d_ws, size_t ws_size,
                              hipStream_t stream)
{
    const float*         x    = (const float*)d_in[0];
    const unsigned char* mask = (const unsigned char*)d_in[1];  // jnp.bool_ -> 1 byte

    const int BTD = in_sizes[0];            // B*T*D
    const int BT  = in_sizes[1];            // B*T
    const int D   = BTD / BT;               // 256
    const int B   = out_size - BTD;         // out = B*T*D floats + B lens
    const int T   = BT / B;                 // 4096
    const int D4  = D / 4;                  // float4s per row

    // Workspace layout: [0,64) int lens[B]; [256, 256+BT*4) int srcIdx.
    int* lensI  = (int*)d_ws;
    int* srcIdx = (int*)((char*)d_ws + 256);

    float* outF    = (float*)d_out;
    float* lensOut = outF + (size_t)BTD;    // lens tail of d_out

    // 1) per-batch mask prefix scan (B blocks of 1024 threads)
    pms_scan_kernel<<<B, 1024, 0, stream>>>(mask, srcIdx, lensI, lensOut, T);

    // 2) gather + pad (one thread per output float4)
    const unsigned int total4 = (unsigned int)((size_t)BTD / 4);
    const unsigned int grid   = (total4 + 255u) / 256u;

    const bool pow2 = ((D4 & (D4 - 1)) == 0) && ((T & (T - 1)) == 0);
    if (pow2) {
        const unsigned int shiftD4 = ilog2u((unsigned int)D4);
        const unsigned int shiftT  = ilog2u((unsigned int)T);
        pms_gather_pow2<<<grid, 256, 0, stream>>>(
            (const float4v*)x, srcIdx, lensI, (float4v*)d_out,
            shiftD4, (unsigned int)(D4 - 1), shiftT, (unsigned int)(T - 1), total4);
    } else {
        pms_gather_div<<<grid, 256, 0, stream>>>(
            (const float4v*)x, srcIdx, lensI, (float4v*)d_out,
            (unsigned int)T, (unsigned int)D4, total4);
    }
}